// GNNEncoder_17059610100266
// MI455X (gfx1250) — compile-verified
//
#include <hip/hip_runtime.h>

// CDNA5 / gfx1250, wave32. All matrix math uses V_WMMA_F32_16X16X4_F32
// (exact fp32 — top-K neighbor selection is discrete and precision-sensitive,
// so we do not downconvert operands).
//
// fp32 WMMA operand layouts used below (ISA 7.12.2):
//   A (16x4):  lane l -> M = l%16, holds K = k0 + 2*(l/16) + {0,1}   (v2f)
//   B (4x16):  lane l -> N = l%16, holds K = k0 + 2*(l/16) + {0,1}   (v2f)
//   C/D:       VGPR r, lane l -> (M = r + 8*(l/16), N = l%16)        (v8f)

typedef __attribute__((ext_vector_type(2))) float v2f;
typedef __attribute__((ext_vector_type(8))) float v8f;

#define BB   32
#define NN   1024
#define KNN  12
#define HD   64

static __device__ __forceinline__ v8f wmma4(v2f a, v2f b, v8f c) {
  return __builtin_amdgcn_wmma_f32_16x16x4_f32(false, a, false, b, (short)0, c,
                                               false, false);
}

// ---------------------------------------------------------------------------
// Layer 0 prep: zero-pad channels 3 -> 4 (so GEMM loads are unconditional,
// 8B-aligned) and compute squared norms.
// ---------------------------------------------------------------------------
__global__ void pad0_kernel(const float* __restrict__ x, float* __restrict__ xpad,
                            float* __restrict__ d2, int total) {
  int t = blockIdx.x * blockDim.x + threadIdx.x;
  if (t < total) {
    float a = x[(size_t)t * 3 + 0];
    float b = x[(size_t)t * 3 + 1];
    float c = x[(size_t)t * 3 + 2];
    xpad[(size_t)t * 4 + 0] = a;
    xpad[(size_t)t * 4 + 1] = b;
    xpad[(size_t)t * 4 + 2] = c;
    xpad[(size_t)t * 4 + 3] = 0.f;
    d2[t] = a * a + b * b + c * c;
  }
}

// Hidden layers: squared norms only (buffer already dense stride-64).
__global__ void d2_kernel(const float* __restrict__ x, float* __restrict__ d2,
                          int total) {
  int t = blockIdx.x * blockDim.x + threadIdx.x;
  if (t < total) {
    float s = 0.f;
#pragma unroll
    for (int c = 0; c < HD; ++c) { float v = x[(size_t)t * HD + c]; s += v * v; }
    d2[t] = s;
  }
}

// ---------------------------------------------------------------------------
// kNN: one block = 16 query rows of one batch. WMMA builds the 16x1024
// distance slab in LDS; then top-12 per row via wave32 shuffle argmin.
// LDS: 64KB slab + A tile (CDNA5 WGP LDS = 320KB).
// ---------------------------------------------------------------------------
template <int CPAD>
__global__ __launch_bounds__(256) void knn_kernel(const float* __restrict__ xpad,
                                                  const float* __restrict__ d2,
                                                  int* __restrict__ nbr) {
  __shared__ float sdist[16][NN];       // 64 KB
  __shared__ float sA[16][CPAD];        // query rows (pre-padded)
  const int b    = blockIdx.y;
  const int i0   = blockIdx.x * 16;
  const int tid  = threadIdx.x;
  const int wave = tid >> 5;
  const int lane = tid & 31;
  const int lg   = lane >> 4;
  const int lm   = lane & 15;

  for (int t = tid; t < 16 * CPAD; t += 256) {
    int r = t / CPAD, c = t % CPAD;
    sA[r][c] = xpad[((size_t)b * NN + i0 + r) * CPAD + c];
  }
  __syncthreads();

  const size_t d2base = (size_t)b * NN;
  float d2i[8];
#pragma unroll
  for (int r = 0; r < 8; ++r) d2i[r] = d2[d2base + i0 + r + 8 * lg];

  const v8f vz = {0.f, 0.f, 0.f, 0.f, 0.f, 0.f, 0.f, 0.f};
  for (int jt = 0; jt < 8; ++jt) {            // each wave owns 128 columns
    const int j0 = (wave * 8 + jt) * 16;
    const float* xj = &xpad[((size_t)b * NN + j0 + lm) * CPAD];
    v8f acc = vz;
#pragma unroll
    for (int ks = 0; ks < CPAD / 4; ++ks) {
      const int ka = ks * 4 + 2 * lg;
      v2f a  = *(const v2f*)&sA[lm][ka];      // ds_load_b64
      v2f bf = *(const v2f*)&xj[ka];          // global_load_b64
      acc = wmma4(a, bf, acc);
    }
    const float d2j = d2[d2base + j0 + lm];
#pragma unroll
    for (int r = 0; r < 8; ++r) {
      const int m  = r + 8 * lg;
      const int gi = i0 + m, gj = j0 + lm;
      float dv = d2i[r] + d2j - 2.f * acc[r];
      if (gi == gj) dv = -1e30f;              // self always selected first
      sdist[m][gj] = dv;
    }
  }
  __syncthreads();

  // top-K: 8 waves x 2 rows each; 12 rounds of argmin-and-knockout
  for (int rr = 0; rr < 2; ++rr) {
    const int m = wave * 2 + rr;
    for (int kk = 0; kk < KNN; ++kk) {
      float best = 1e38f; int bi = NN;
      for (int j = lane; j < NN; j += 32) {
        float v = sdist[m][j];
        if (v < best) { best = v; bi = j; }
      }
#pragma unroll
      for (int off = 16; off >= 1; off >>= 1) {
        float ov = __shfl_down(best, off, 32);
        int   oi = __shfl_down(bi,   off, 32);
        if (ov < best || (ov == best && oi < bi)) { best = ov; bi = oi; }
      }
      if (lane == 0) {
        nbr[((size_t)b * NN + i0 + m) * KNN + kk] = bi;
        sdist[m][bi] = 1e38f;                 // knockout; DS in-order per wave
      }
    }
  }
}

// ---------------------------------------------------------------------------
// Edge MLP. Block = 6 waves = 96 edges = 8 nodes. Each wave owns a 16-edge
// M-tile and runs both GEMMs with WMMA (W1/W2 held transposed in LDS so both
// A and B fragments are contiguous v2f loads).
// pass 1: accumulate BN stats (LDS ds_add_f32 + one global atomic per block).
// pass 2: folded BN affine + ReLU, GEMM2, mean over K per node.
// ---------------------------------------------------------------------------
template <int CIN, int CPAD>
__global__ __launch_bounds__(192) void edge_mlp_kernel(
    const float* __restrict__ xpad, const int* __restrict__ nbr,
    const float* __restrict__ W1, const float* __restrict__ b1,
    const float* __restrict__ aff,                 // [scale[64], shift[64]]
    const float* __restrict__ W2, const float* __restrict__ b2,
    float* __restrict__ out,                       // [B,N,64] (pass 2)
    float* __restrict__ stats,                     // [sum[64], sumsq[64]] (pass 1)
    int pass) {
  constexpr int CK2 = 2 * CPAD;         // padded edge-feature width (8 or 128)
  __shared__ float sE[6][16][CK2];      // per-wave edge features
  __shared__ float sW1t[HD][CK2];       // W1 transposed: [h][k]
  __shared__ float sW2t[HD][HD];        // W2 transposed: [h][k]
  __shared__ float sH[96][HD];          // h1n, then overwritten with h2
  __shared__ float sSum[HD], sSq[HD];

  const int tid  = threadIdx.x;
  const int wave = tid >> 5;
  const int lane = tid & 31;
  const int lg   = lane >> 4;
  const int lm   = lane & 15;

  for (int t = tid; t < HD * CK2; t += 192) {
    int h = t / CK2, s = t % CK2;
    int half = (s >= CPAD) ? 1 : 0;
    int o = s - half * CPAD;
    sW1t[h][s] = (o < CIN) ? W1[(half * CIN + o) * HD + h] : 0.f;
  }
  if (pass == 2)
    for (int t = tid; t < HD * HD; t += 192) sW2t[t / HD][t % HD] = W2[(t % HD) * HD + (t / HD)];
  if (pass == 1 && tid < HD) { sSum[tid] = 0.f; sSq[tid] = 0.f; }

  // gather this wave's 16 edges: e = [xi_pad , xj_pad - xi_pad]
  const int e0 = blockIdx.x * 96 + wave * 16;     // global edge base
  for (int r = 0; r < 16; ++r) {
    const int ge = e0 + r;
    const int gn = ge / KNN;                      // flattened (b*N+n)
    const int j  = nbr[ge];
    const int bb = gn / NN;
    const float* xi = &xpad[(size_t)gn * CPAD];
    const float* xj = &xpad[((size_t)bb * NN + j) * CPAD];
    for (int s = lane; s < CK2; s += 32) {
      float v;
      if (s < CPAD) v = xi[s];
      else          v = xj[s - CPAD] - xi[s - CPAD];
      sE[wave][r][s] = v;
    }
  }
  __syncthreads();

  // GEMM1: [16 x CK2] @ [CK2 x 64]
  v8f acc[4];
  const v8f vz = {0.f, 0.f, 0.f, 0.f, 0.f, 0.f, 0.f, 0.f};
#pragma unroll
  for (int nt = 0; nt < 4; ++nt) acc[nt] = vz;
#pragma unroll
  for (int k0 = 0; k0 < CK2; k0 += 4) {
    const int ka = k0 + 2 * lg;
    v2f a = *(const v2f*)&sE[wave][lm][ka];
#pragma unroll
    for (int nt = 0; nt < 4; ++nt) {
      v2f bf = *(const v2f*)&sW1t[nt * 16 + lm][ka];
      acc[nt] = wmma4(a, bf, acc[nt]);
    }
  }

  if (pass == 1) {
#pragma unroll
    for (int nt = 0; nt < 4; ++nt) {
      const int f = nt * 16 + lm;
      const float bias = b1[f];
      float s = 0.f, q = 0.f;
#pragma unroll
      for (int r = 0; r < 8; ++r) {
        float v = acc[nt][r] + bias;
        s += v; q += v * v;
      }
      atomicAdd(&sSum[f], s);     // ds_add_f32
      atomicAdd(&sSq[f],  q);
    }
    __syncthreads();
    if (tid < HD) {
      atomicAdd(&stats[tid],      sSum[tid]);
      atomicAdd(&stats[HD + tid], sSq[tid]);
    }
    return;
  }

  // pass 2: folded BN affine + ReLU -> stage h1n
#pragma unroll
  for (int nt = 0; nt < 4; ++nt) {
    const int f = nt * 16 + lm;
    const float bias = b1[f], sc = aff[f], sh = aff[HD + f];
#pragma unroll
    for (int r = 0; r < 8; ++r) {
      const int m = r + 8 * lg;
      float v = fmaf(acc[nt][r] + bias, sc, sh);
      sH[wave * 16 + m][f] = v > 0.f ? v : 0.f;
    }
  }
  // GEMM2: [16 x 64] @ [64 x 64]  (reads own wave's rows; DS in-order per wave)
  v8f acc2[4];
#pragma unroll
  for (int ot = 0; ot < 4; ++ot) acc2[ot] = vz;
#pragma unroll
  for (int k0 = 0; k0 < HD; k0 += 4) {
    const int ka = k0 + 2 * lg;
    v2f a = *(const v2f*)&sH[wave * 16 + lm][ka];
#pragma unroll
    for (int ot = 0; ot < 4; ++ot) {
      v2f bf = *(const v2f*)&sW2t[ot * 16 + lm][ka];
      acc2[ot] = wmma4(a, bf, acc2[ot]);
    }
  }
  // overwrite staging with h2 = h1n@W2 + b2 (all reads retired into acc2)
#pragma unroll
  for (int ot = 0; ot < 4; ++ot) {
    const int f = ot * 16 + lm;
    const float bias = b2[f];
#pragma unroll
    for (int r = 0; r < 8; ++r) {
      const int m = r + 8 * lg;
      sH[wave * 16 + m][f] = acc2[ot][r] + bias;
    }
  }
  __syncthreads();

  // mean over K=12 per node (8 nodes per block)
  for (int t = tid; t < 8 * HD; t += 192) {
    const int nl = t / HD, f = t % HD;
    float s = 0.f;
#pragma unroll
    for (int k = 0; k < KNN; ++k) s += sH[nl * 12 + k][f];
    const int gn = blockIdx.x * 8 + nl;
    out[(size_t)gn * HD + f] = s * (1.f / 12.f);
  }
}

// ---------------------------------------------------------------------------
__global__ void zero_kernel(float* p, int n) {
  int t = blockIdx.x * blockDim.x + threadIdx.x;
  if (t < n) p[t] = 0.f;
}

__global__ void bn_finalize_kernel(const float* __restrict__ stats,
                                   const float* __restrict__ g,
                                   const float* __restrict__ be,
                                   float* __restrict__ aff, float inv_cnt) {
  int f = threadIdx.x;
  if (f < HD) {
    float m  = stats[f] * inv_cnt;
    float v  = stats[HD + f] * inv_cnt - m * m;   // biased variance
    float is = rsqrtf(v + 1e-5f);
    float sc = g[f] * is;
    aff[f]      = sc;
    aff[HD + f] = be[f] - sc * m;
  }
}

__global__ void pool_kernel(const float* __restrict__ x, float* __restrict__ pooled) {
  int t = blockIdx.x * blockDim.x + threadIdx.x;   // B*HD threads
  if (t < BB * HD) {
    int b = t / HD, f = t % HD;
    float s = 0.f;
    for (int n = 0; n < NN; ++n) s += x[((size_t)b * NN + n) * HD + f];
    pooled[t] = s * (1.f / NN);
  }
}

// Final head: fc1 (64->32) + training-mode BN over batch + ReLU + fc2 (32->2)
__global__ __launch_bounds__(256) void head_kernel(
    const float* __restrict__ pooled,
    const float* __restrict__ wf1, const float* __restrict__ bf1,
    const float* __restrict__ gf,  const float* __restrict__ bef,
    const float* __restrict__ wf2, const float* __restrict__ bf2,
    float* __restrict__ out) {
  __shared__ float z[BB][32];
  __shared__ float ssc[32], ssh[32];
  const int tid = threadIdx.x;
  for (int t = tid; t < BB * 32; t += 256) {
    int b = t / 32, f = t % 32;
    float s = bf1[f];
#pragma unroll
    for (int c = 0; c < HD; ++c) s += pooled[b * HD + c] * wf1[c * 32 + f];
    z[b][f] = s;
  }
  __syncthreads();
  if (tid < 32) {
    float s = 0.f, q = 0.f;
#pragma unroll
    for (int b = 0; b < BB; ++b) { float v = z[b][tid]; s += v; q += v * v; }
    float m  = s * (1.f / BB);
    float va = q * (1.f / BB) - m * m;
    float is = rsqrtf(va + 1e-5f);
    ssc[tid] = gf[tid] * is;
    ssh[tid] = bef[tid] - gf[tid] * is * m;
  }
  __syncthreads();
  for (int t = tid; t < BB * 32; t += 256) {
    int b = t / 32, f = t % 32;
    float v = fmaf(z[b][f], ssc[f], ssh[f]);
    z[b][f] = v > 0.f ? v : 0.f;
  }
  __syncthreads();
  for (int t = tid; t < BB * 2; t += 256) {
    int b = t / 2, o = t % 2;
    float s = bf2[o];
#pragma unroll
    for (int c = 0; c < 32; ++c) s += z[b][c] * wf2[c * 2 + o];
    out[b * 2 + o] = s;
  }
}

// ---------------------------------------------------------------------------
extern "C" void kernel_launch(void* const* d_in, const int* in_sizes, int n_in,
                              void* d_out, int out_size, void* d_ws, size_t ws_size,
                              hipStream_t stream) {
  const float* x    = (const float*)d_in[0];
  const float* w0a  = (const float*)d_in[1];
  const float* b0a  = (const float*)d_in[2];
  const float* g0   = (const float*)d_in[3];
  const float* be0  = (const float*)d_in[4];
  const float* w0b  = (const float*)d_in[5];
  const float* b0b  = (const float*)d_in[6];
  const float* wa   = (const float*)d_in[7];
  const float* ba   = (const float*)d_in[8];
  const float* ga   = (const float*)d_in[9];
  const float* bea  = (const float*)d_in[10];
  const float* wb   = (const float*)d_in[11];
  const float* bbp  = (const float*)d_in[12];
  const float* wf1  = (const float*)d_in[13];
  const float* bf1  = (const float*)d_in[14];
  const float* gf   = (const float*)d_in[15];
  const float* bef  = (const float*)d_in[16];
  const float* wf2  = (const float*)d_in[17];
  const float* bf2  = (const float*)d_in[18];
  (void)in_sizes; (void)n_in; (void)out_size; (void)ws_size;

  float* wsf    = (float*)d_ws;
  float* xb0    = wsf;                                    // [B,N,64]
  float* xb1    = xb0 + (size_t)BB * NN * HD;             // [B,N,64]
  float* xp4    = xb1 + (size_t)BB * NN * HD;             // [B,N,4] layer-0 pad
  float* d2b    = xp4 + (size_t)BB * NN * 4;              // [B,N]
  int*   nbrb   = (int*)(d2b + (size_t)BB * NN);          // [B,N,K]
  float* stats  = (float*)(nbrb + (size_t)BB * NN * KNN); // 128
  float* aff    = stats + 128;                            // 128
  float* pooled = aff + 128;                              // B*64

  const float inv_cnt = 1.f / ((float)BB * NN * KNN);
  const int total = BB * NN;

  // ---- layer 0 (Cin = 3, padded to 4) ----
  pad0_kernel<<<(total + 255) / 256, 256, 0, stream>>>(x, xp4, d2b, total);
  knn_kernel<4><<<dim3(NN / 16, BB), 256, 0, stream>>>(xp4, d2b, nbrb);
  zero_kernel<<<1, 128, 0, stream>>>(stats, 128);
  edge_mlp_kernel<3, 4><<<BB * NN / 8, 192, 0, stream>>>(
      xp4, nbrb, w0a, b0a, nullptr, nullptr, nullptr, nullptr, stats, 1);
  bn_finalize_kernel<<<1, 64, 0, stream>>>(stats, g0, be0, aff, inv_cnt);
  edge_mlp_kernel<3, 4><<<BB * NN / 8, 192, 0, stream>>>(
      xp4, nbrb, w0a, b0a, aff, w0b, b0b, xb0, nullptr, 2);

  // ---- layers 1..3 (Cin = 64, dense) ----
  const float* cur = xb0;
  for (int l = 1; l < 4; ++l) {
    const float* W1 = wa  + (size_t)(l - 1) * 128 * HD;
    const float* B1 = ba  + (size_t)(l - 1) * HD;
    const float* G  = ga  + (size_t)(l - 1) * HD;
    const float* BE = bea + (size_t)(l - 1) * HD;
    const float* W2 = wb  + (size_t)(l - 1) * HD * HD;
    const float* B2 = bbp + (size_t)(l - 1) * HD;

    d2_kernel<<<(total + 255) / 256, 256, 0, stream>>>(cur, d2b, total);
    knn_kernel<64><<<dim3(NN / 16, BB), 256, 0, stream>>>(cur, d2b, nbrb);
    zero_kernel<<<1, 128, 0, stream>>>(stats, 128);
    edge_mlp_kernel<64, 64><<<BB * NN / 8, 192, 0, stream>>>(
        cur, nbrb, W1, B1, nullptr, nullptr, nullptr, nullptr, stats, 1);
    bn_finalize_kernel<<<1, 64, 0, stream>>>(stats, G, BE, aff, inv_cnt);
    float* nxt = (l & 1) ? xb1 : xb0;
    edge_mlp_kernel<64, 64><<<BB * NN / 8, 192, 0, stream>>>(
        cur, nbrb, W1, B1, aff, W2, B2, nxt, nullptr, 2);
    cur = nxt;
  }

  pool_kernel<<<(BB * HD + 255) / 256, 256, 0, stream>>>(cur, pooled);
  head_kernel<<<1, 256, 0, stream>>>(pooled, wf1, bf1, gf, bef, wf2, bf2,
                                     (float*)d_out);
}